// GNN_18391049961554
// MI455X (gfx1250) — compile-verified
//
#include <hip/hip_runtime.h>

// ---------------------------------------------------------------------------
// GNN diffusion forward on gfx1250 (MI455X).
// Dense GEMMs on V_WMMA_F32_16X16X4_F32 (f32 exact, matches reference).
// Edge gather/scatter as wave-per-edge float4 + global_atomic_add_f32
// (node buffers are 51.2MB each -> fully L2 resident on 192MB L2).
// ---------------------------------------------------------------------------

#define N_FEAT   64
#define HID      128
#define N_GRAPHS 512
#define N_TGT    16

typedef __attribute__((ext_vector_type(2))) float v2f;
typedef __attribute__((ext_vector_type(8))) float v8f;

__device__ __forceinline__ void atomAdd(float* p, float v) {
    __hip_atomic_fetch_add(p, v, __ATOMIC_RELAXED, __HIP_MEMORY_SCOPE_AGENT);
}

// ------------------------------ small utility kernels ----------------------

__global__ __launch_bounds__(256) void fill_f32(float* __restrict__ p, float v, int n) {
    int i = blockIdx.x * 256 + threadIdx.x;
    if (i < n) p[i] = v;
}

__global__ __launch_bounds__(256) void deg_count(const int* __restrict__ dst,
                                                 float* __restrict__ deg, int E) {
    int e = blockIdx.x * 256 + threadIdx.x;
    if (e < E) atomAdd(&deg[dst[e]], 1.0f);
}

__global__ __launch_bounds__(256) void rsqrt_ip(float* __restrict__ d, int n) {
    int i = blockIdx.x * 256 + threadIdx.x;
    if (i < n) d[i] = rsqrtf(d[i]);   // deg >= 1 (self loop baked in)
}

// ------------------------------ WMMA f32 GEMM ------------------------------
// Y[M,128] = X[M,K] @ W[K,128]  (+bias, +relu optional)
// M % 16 == 0 (100000 = 6250*16). Block: 8 waves; wave w -> N-tile w.
template <int K, bool RELU, bool BIAS>
__global__ __launch_bounds__(256) void gemm_wmma_f32(const float* __restrict__ X,
                                                     const float* __restrict__ W,
                                                     const float* __restrict__ bias,
                                                     float* __restrict__ Y) {
    const int wave  = threadIdx.x >> 5;
    const int lane  = threadIdx.x & 31;
    const int hi    = lane >> 4;      // lane half selects K pair
    const int lr    = lane & 15;
    const int mBase = blockIdx.x << 4;
    const int nBase = wave << 4;

    // A 16x4 f32 fragment: lanes 0-15 rows, VGPR0/1 = K {0,1} (half 0) or {2,3} (half 1)
    const float* __restrict__ arow = X + (size_t)(mBase + lr) * K + 2 * hi;
    // B 4x16 f32 fragment: mirrored layout, col = lr
    const float* __restrict__ bcol = W + (size_t)(2 * hi) * HID + nBase + lr;

    v8f c = {};
#pragma unroll
    for (int k0 = 0; k0 < K; k0 += 4) {
        v2f a, b;
        a.x = arow[k0];
        a.y = arow[k0 + 1];
        b.x = bcol[(size_t)k0 * HID];
        b.y = bcol[(size_t)k0 * HID + HID];
        c = __builtin_amdgcn_wmma_f32_16x16x4_f32(false, a, false, b,
                                                  (short)0, c, false, false);
    }

    const float bv = BIAS ? bias[nBase + lr] : 0.0f;
    float* __restrict__ yout = Y + (size_t)mBase * HID + nBase + lr;
#pragma unroll
    for (int j = 0; j < 8; ++j) {     // C/D layout: VGPR j -> row j (+8 for lane half 1)
        float v = c[j] + bv;
        if (RELU) v = fmaxf(v, 0.0f);
        yout[(size_t)(j + 8 * hi) * HID] = v;
    }
}

// ------------------------------ GCN aggregation ----------------------------

// agg[i,:] = dinv[i]^2 * hw[i,:]   (self-loop term; also initializes agg)
__global__ __launch_bounds__(256) void agg_selfloop(const float* __restrict__ hw,
                                                    const float* __restrict__ dinv,
                                                    float* __restrict__ agg, int total) {
    int i = blockIdx.x * 256 + threadIdx.x;
    if (i < total) {
        float dv = dinv[i >> 7];
        agg[i] = dv * dv * hw[i];
    }
}

// one wave per edge: agg[dst,:] += dinv[src]*dinv[dst] * hw[src,:]
__global__ __launch_bounds__(256) void edge_scatter(const int* __restrict__ src,
                                                    const int* __restrict__ dst,
                                                    const float* __restrict__ dinv,
                                                    const float* __restrict__ hw,
                                                    float* __restrict__ agg, int E) {
    int e = blockIdx.x * 8 + (threadIdx.x >> 5);
    if (e >= E) return;
    int lane = threadIdx.x & 31;
    int s = src[e], d = dst[e];
    float coef = dinv[s] * dinv[d];
    const float4 v = ((const float4*)(hw + (size_t)s * HID))[lane];
    float* ad = agg + (size_t)d * HID + (lane << 2);
    atomAdd(ad + 0, coef * v.x);
    atomAdd(ad + 1, coef * v.y);
    atomAdd(ad + 2, coef * v.z);
    atomAdd(ad + 3, coef * v.w);
}

// out = relu(agg + b) + temb     (layer 1 epilogue)
__global__ __launch_bounds__(256) void bias_relu_add(const float* __restrict__ agg,
                                                     const float* __restrict__ b,
                                                     const float* __restrict__ temb,
                                                     float* __restrict__ out, int total) {
    int i = blockIdx.x * 256 + threadIdx.x;
    if (i < total) out[i] = fmaxf(agg[i] + b[i & (HID - 1)], 0.0f) + temb[i];
}

// out = relu(agg + b)
__global__ __launch_bounds__(256) void bias_relu(const float* __restrict__ agg,
                                                 const float* __restrict__ b,
                                                 float* __restrict__ out, int total) {
    int i = blockIdx.x * 256 + threadIdx.x;
    if (i < total) out[i] = fmaxf(agg[i] + b[i & (HID - 1)], 0.0f);
}

// ------------------------------ pooling + head -----------------------------

__global__ __launch_bounds__(256) void pool_scatter(const float* __restrict__ h,
                                                    const int* __restrict__ batch,
                                                    float* __restrict__ gsum,
                                                    float* __restrict__ cnt, int N) {
    int n = blockIdx.x * 8 + (threadIdx.x >> 5);
    if (n >= N) return;
    int lane = threadIdx.x & 31;
    int g = batch[n];
    const float4 v = ((const float4*)(h + (size_t)n * HID))[lane];
    float* gs = gsum + (size_t)g * HID + (lane << 2);
    atomAdd(gs + 0, v.x);
    atomAdd(gs + 1, v.y);
    atomAdd(gs + 2, v.z);
    atomAdd(gs + 3, v.w);
    if (lane == 0) atomAdd(&cnt[g], 1.0f);
}

__global__ __launch_bounds__(256) void head_out(const float* __restrict__ gsum,
                                                const float* __restrict__ cnt,
                                                const float* __restrict__ Wo,
                                                const float* __restrict__ bo,
                                                float* __restrict__ out) {
    int idx = blockIdx.x * 256 + threadIdx.x;
    if (idx >= N_GRAPHS * N_TGT) return;
    int g = idx / N_TGT, t = idx % N_TGT;
    float inv = 1.0f / fmaxf(cnt[g], 1.0f);
    const float* gv = gsum + (size_t)g * HID;
    float acc = bo[t];
#pragma unroll 8
    for (int k = 0; k < HID; ++k) acc += gv[k] * inv * Wo[k * N_TGT + t];
    out[idx] = acc;
}

// ------------------------------ driver -------------------------------------

extern "C" void kernel_launch(void* const* d_in, const int* in_sizes, int n_in,
                              void* d_out, int out_size, void* d_ws, size_t ws_size,
                              hipStream_t stream) {
    const float* x    = (const float*)d_in[0];
    const int*   eidx = (const int*)d_in[1];
    const float* temb = (const float*)d_in[2];
    const int*   batch = (const int*)d_in[3];
    const float* Wt = (const float*)d_in[4];
    const float* bt = (const float*)d_in[5];
    const float* W1 = (const float*)d_in[6];
    const float* b1 = (const float*)d_in[7];
    const float* Ws = (const float*)d_in[8];
    const float* bs = (const float*)d_in[9];
    const float* Wo = (const float*)d_in[10];
    const float* bo = (const float*)d_in[11];
    float* out = (float*)d_out;

    const int N = in_sizes[0] / N_FEAT;      // 100000
    const int E = in_sizes[1] / 2;           // 1600000
    const int* src = eidx;
    const int* dst = eidx + E;
    const size_t NH = (size_t)N * HID;

    float* ws   = (float*)d_ws;
    float* dinv = ws;                         // N floats
    float* A    = ws + 102400;                // node buffers, N*HID each
    float* B    = A + NH;
    float* C    = B + NH;
    float* gsum = C + NH;                     // 512*128
    float* cnt  = gsum + (size_t)N_GRAPHS * HID;

    dim3 blk(256);
    auto cdiv = [](int a, int b) { return (a + b - 1) / b; };
    const int mTiles = N / 16;
    const int nhB = cdiv((int)NH, 256);

    // degrees (init to 1 for self loop), then dinv = rsqrt(deg)
    fill_f32<<<cdiv(N, 256), blk, 0, stream>>>(dinv, 1.0f, N);
    deg_count<<<cdiv(E, 256), blk, 0, stream>>>(dst, dinv, E);
    rsqrt_ip<<<cdiv(N, 256), blk, 0, stream>>>(dinv, N);

    // t_emb = relu(T @ Wt + bt) -> C
    gemm_wmma_f32<128, true, true><<<mTiles, blk, 0, stream>>>(temb, Wt, bt, C);

    // layer 1: HW = x @ W1 -> A ; agg -> B ; h = relu(agg+b1)+t_emb -> A
    gemm_wmma_f32<64, false, false><<<mTiles, blk, 0, stream>>>(x, W1, nullptr, A);
    agg_selfloop<<<nhB, blk, 0, stream>>>(A, dinv, B, (int)NH);
    edge_scatter<<<cdiv(E, 8), blk, 0, stream>>>(src, dst, dinv, A, B, E);
    bias_relu_add<<<nhB, blk, 0, stream>>>(B, b1, C, A, (int)NH);

    // layers 2..3
    for (int i = 0; i < 2; ++i) {
        gemm_wmma_f32<128, false, false><<<mTiles, blk, 0, stream>>>(
            A, Ws + (size_t)i * HID * HID, nullptr, B);
        agg_selfloop<<<nhB, blk, 0, stream>>>(B, dinv, C, (int)NH);
        edge_scatter<<<cdiv(E, 8), blk, 0, stream>>>(src, dst, dinv, B, C, E);
        bias_relu<<<nhB, blk, 0, stream>>>(C, bs + (size_t)i * HID, A, (int)NH);
    }

    // global mean pool + linear head
    fill_f32<<<cdiv(N_GRAPHS * HID, 256), blk, 0, stream>>>(gsum, 0.0f, N_GRAPHS * HID);
    fill_f32<<<cdiv(N_GRAPHS, 256), blk, 0, stream>>>(cnt, 0.0f, N_GRAPHS);
    pool_scatter<<<cdiv(N, 8), blk, 0, stream>>>(A, batch, gsum, cnt, N);
    head_out<<<cdiv(N_GRAPHS * N_TGT, 256), blk, 0, stream>>>(gsum, cnt, Wo, bo, out);
}